// AdditiveEmission_29540785062656
// MI455X (gfx1250) — compile-verified
//
#include <hip/hip_runtime.h>
#include <math.h>

typedef __attribute__((ext_vector_type(2))) float v2f;
typedef __attribute__((ext_vector_type(8))) float v8f;

#define B_   4
#define L_   512
#define D_   128
#define ROWS (B_ * L_)   // 2048 flattened (b, i) rows

// ---------------------------------------------------------------------------
// Kernel 1: fused Q = X*Wt and K = X*Wx using V_WMMA_F32_16X16X4_F32.
// One wave computes one 16x16 tile of BOTH Q and K (shared A operand).
// M = 2048 (b*L+i), N = 128 (d), K-dim = 128, stepped by 4 per WMMA.
// ---------------------------------------------------------------------------
__global__ __launch_bounds__(128) void qk_gemm_wmma(
    const float* __restrict__ X, const float* __restrict__ Wt,
    const float* __restrict__ Wx, float* __restrict__ Q, float* __restrict__ K)
{
  const int wave = blockIdx.x * (blockDim.x >> 5) + (threadIdx.x >> 5);
  const int lane = threadIdx.x & 31;
  const int r    = lane & 15;   // row within tile (A) / col within tile (B,C,D)
  const int half = lane >> 4;   // selects K=0/1 vs K=2/3 (A,B); M vs M+8 (C,D)

  const int mt = wave >> 3;     // 0..127 M-tiles
  const int nt = wave & 7;      // 0..7   N-tiles
  const int m0 = mt << 4;
  const int n0 = nt << 4;

  v8f accQ = {};
  v8f accK = {};

  // A-matrix 16x4 f32 layout: lanes 0-15 -> {K=0 in v0, K=1 in v1},
  // lanes 16-31 -> {K=2 in v0, K=3 in v1}.  Per-lane float2 is contiguous.
  const float* arow = X + (m0 + r) * D_ + 2 * half;

  for (int k0 = 0; k0 < D_; k0 += 4) {
    v2f a = *(const v2f*)(arow + k0);

    const int krow = k0 + 2 * half;     // B-matrix 4x16: mirrored layout
    v2f bq, bk;
    bq.x = Wt[(krow    ) * D_ + n0 + r];
    bq.y = Wt[(krow + 1) * D_ + n0 + r];
    bk.x = Wx[(krow    ) * D_ + n0 + r];
    bk.y = Wx[(krow + 1) * D_ + n0 + r];

    // (neg_a, A, neg_b, B, c_mod, C, reuse_a, reuse_b)
    accQ = __builtin_amdgcn_wmma_f32_16x16x4_f32(
        false, a, false, bq, (short)0, accQ, false, false);
    accK = __builtin_amdgcn_wmma_f32_16x16x4_f32(
        false, a, false, bk, (short)0, accK, false, false);
  }

  // C/D 16x16 f32 layout: VGPR v, lanes 0-15 -> M=v, lanes 16-31 -> M=v+8.
  #pragma unroll
  for (int v = 0; v < 8; ++v) {
    const int row = m0 + v + 8 * half;
    const int col = n0 + r;
    Q[row * D_ + col] = accQ[v];
    K[row * D_ + col] = accK[v];
  }
}

// ---------------------------------------------------------------------------
// Kernel 2: one wave32 per (b, i) query row.
// Each lane owns 4 contiguous d's (float4).  Window j in [i-1, i+2):
//   e_j = sum_d Wa[d]*tanh(q[i,d]+k[j,d]+bh[d]) + ba
// stable softmax over the window (deviation from reference's global max is
// bounded by EPS-level terms, << f32 noise), then out = sum_j a_j * X[b,j,:].
// ---------------------------------------------------------------------------
__global__ __launch_bounds__(256) void attn_window(
    const float* __restrict__ X, const float* __restrict__ Q,
    const float* __restrict__ K, const float* __restrict__ Wa,
    const float* __restrict__ bh, const float* __restrict__ ba,
    float* __restrict__ out)
{
  const int wave = blockIdx.x * (blockDim.x >> 5) + (threadIdx.x >> 5);
  const int lane = threadIdx.x & 31;
  const int b = wave / L_;
  const int i = wave % L_;
  const int d0 = lane << 2;

  const float4 qv  = *(const float4*)(Q  + wave * D_ + d0);
  const float4 bhv = *(const float4*)(bh + d0);
  const float4 wav = *(const float4*)(Wa + d0);
  const float  bav = ba[0];

  bool  valid[3];
  float e[3];
  #pragma unroll
  for (int jj = 0; jj < 3; ++jj) {
    const int j = i - 1 + jj;           // lower = i - ATTN_WIDTH/2
    valid[jj] = (j >= 0) && (j < L_);
    float p = -3.0e38f;
    if (valid[jj]) {                    // wave-uniform branch (same i per wave)
      const float4 kv = *(const float4*)(K + (b * L_ + j) * D_ + d0);
      float s = wav.x * tanhf(qv.x + kv.x + bhv.x)
              + wav.y * tanhf(qv.y + kv.y + bhv.y)
              + wav.z * tanhf(qv.z + kv.z + bhv.z)
              + wav.w * tanhf(qv.w + kv.w + bhv.w);
      #pragma unroll
      for (int off = 16; off > 0; off >>= 1) s += __shfl_xor(s, off, 32);
      p = s + bav;
    }
    e[jj] = p;
  }

  const float m = fmaxf(e[0], fmaxf(e[1], e[2]));
  float w[3], S = 0.0f;
  #pragma unroll
  for (int jj = 0; jj < 3; ++jj) {
    w[jj] = valid[jj] ? expf(e[jj] - m) : 0.0f;
    S += w[jj];
  }
  const float inv = 1.0f / (S + 1e-8f);

  float4 o = make_float4(0.f, 0.f, 0.f, 0.f);
  #pragma unroll
  for (int jj = 0; jj < 3; ++jj) {
    const int j = i - 1 + jj;
    if (valid[jj]) {
      const float a = w[jj] * inv;
      const float4 xv = *(const float4*)(X + (b * L_ + j) * D_ + d0);
      o.x += a * xv.x;  o.y += a * xv.y;
      o.z += a * xv.z;  o.w += a * xv.w;
    }
  }
  *(float4*)(out + wave * D_ + d0) = o;
}

// ---------------------------------------------------------------------------
extern "C" void kernel_launch(void* const* d_in, const int* in_sizes, int n_in,
                              void* d_out, int out_size, void* d_ws, size_t ws_size,
                              hipStream_t stream) {
  const float* X  = (const float*)d_in[0];  // [B, L, D]
  const float* Wt = (const float*)d_in[1];  // [D, D]
  const float* Wx = (const float*)d_in[2];  // [D, D]
  const float* Wa = (const float*)d_in[3];  // [D, 1]
  const float* bh = (const float*)d_in[4];  // [D]
  const float* ba = (const float*)d_in[5];  // [1]
  float* out = (float*)d_out;               // [B, L, D]

  float* Q = (float*)d_ws;                  // ROWS * D_ floats (1 MB)
  float* K = Q + (size_t)ROWS * D_;         // ROWS * D_ floats (1 MB)

  // 128 M-tiles x 8 N-tiles = 1024 waves; 4 waves/block -> 256 blocks
  qk_gemm_wmma<<<256, 128, 0, stream>>>(X, Wt, Wx, Q, K);
  // 2048 (b,i) rows, 8 waves/block -> 256 blocks
  attn_window<<<256, 256, 0, stream>>>(X, Q, K, Wa, bh, ba, out);
}